// RGCN_36429912605244
// MI455X (gfx1250) — compile-verified
//
#include <hip/hip_runtime.h>

// ---------------------------------------------------------------------------
// RGCN on MI455X (gfx1250): fp32 WMMA GEMMs + atomic edge scatter.
// D = A x B + C via V_WMMA_F32_16X16X4_F32 (wave32, 16x16 tile per wave).
// W is staged into LDS with CDNA5 async global->LDS copies (ASYNCcnt).
// Row scaling (D^-1/2) is commuted to the epilogue: diag(s)*(A@W).
// ---------------------------------------------------------------------------

typedef __attribute__((ext_vector_type(2))) float v2f;
typedef __attribute__((ext_vector_type(8))) float v8f;

#define HID 128

// ---------------- zero fill ----------------
__global__ void rgcn_zero_f32(float* __restrict__ p, long n) {
    long i = (long)blockIdx.x * blockDim.x + threadIdx.x;
    long stride = (long)gridDim.x * blockDim.x;
    for (; i < n; i += stride) p[i] = 0.0f;
}

// ---------------- degree histograms (u32 atomics, native) ----------------
__global__ void rgcn_degrees(const int* __restrict__ su, const int* __restrict__ du,
                             const int* __restrict__ si, const int* __restrict__ di,
                             unsigned* __restrict__ dOU, unsigned* __restrict__ dIU,
                             unsigned* __restrict__ dOI, unsigned* __restrict__ dII,
                             int n) {
    int i = blockIdx.x * blockDim.x + threadIdx.x;
    if (i < n) {
        atomicAdd(&dOU[su[i]], 1u);
        atomicAdd(&dIU[du[i]], 1u);
        atomicAdd(&dOI[si[i]], 1u);
        atomicAdd(&dII[di[i]], 1u);
    }
}

// ---------------- counts -> rsqrt(max(deg,1)) in place ----------------
__global__ void rgcn_scales(float* __restrict__ a, float* __restrict__ b,
                            float* __restrict__ c, float* __restrict__ d, int n) {
    int i = blockIdx.x * blockDim.x + threadIdx.x;
    if (i < n) {
        unsigned ca = ((const unsigned*)a)[i];
        unsigned cb = ((const unsigned*)b)[i];
        unsigned cc = ((const unsigned*)c)[i];
        unsigned cd = ((const unsigned*)d)[i];
        a[i] = rsqrtf(fmaxf((float)ca, 1.0f));
        b[i] = rsqrtf(fmaxf((float)cb, 1.0f));
        c[i] = rsqrtf(fmaxf((float)cc, 1.0f));
        d[i] = rsqrtf(fmaxf((float)cd, 1.0f));
    }
}

// ---------------- WMMA GEMM: C[Mx128] = diag(rowScale) * (A @ W) (+bias) ----
// 256 threads = 8 waves; wave w computes a 16x16 tile at cols [16w, 16w+16).
// Persistent blocks stride over row tiles so W is staged into LDS once per
// block instead of once per tile. M % 16 == 0 (50000 = 16*3125) -> EXEC is
// all-ones through every WMMA, as the ISA requires.
__global__ __launch_bounds__(256) void rgcn_gemm_wmma(
    const float* __restrict__ A, const float* __restrict__ W,
    const float* __restrict__ rowScale,   // nullptr -> 1.0
    const float* __restrict__ bias,       // nullptr -> 0.0
    float* __restrict__ C, int numTiles) {
    __shared__ float sW[HID * HID];       // 64 KB of the 320 KB WGP LDS

    // Stage W into LDS with async global->LDS b128 copies (ASYNCcnt-tracked).
    // Generic pointers into the LDS aperture carry the LDS byte address in
    // their low 32 bits (flat->LDS truncation, ISA 10.2).
    for (int i = threadIdx.x; i < (HID * HID) / 4; i += 256) {
        unsigned lds_off = (unsigned)(uintptr_t)(&sW[i * 4]);
        const float* gp = W + (size_t)i * 4;
        asm volatile("global_load_async_to_lds_b128 %0, %1, off"
                     :: "v"(lds_off), "v"(gp) : "memory");
    }
    asm volatile("s_wait_asynccnt 0" ::: "memory");
    __syncthreads();

    const int wave = threadIdx.x >> 5;
    const int lane = threadIdx.x & 31;
    const int g    = lane >> 4;   // half-wave: selects K pair {0,1} vs {2,3}
    const int r    = lane & 15;
    const int col  = wave * 16 + r;              // B/D column owned by this lane
    const float badd = bias ? bias[col] : 0.0f;  // invariant across tiles

    for (int tile = blockIdx.x; tile < numTiles; tile += gridDim.x) {
        const int rowBase = tile * 16;
        const float* __restrict__ arow = A + (size_t)(rowBase + r) * HID + 2 * g;

        v8f acc = {};
        #pragma unroll
        for (int k = 0; k < HID; k += 4) {
            // A fragment 16x4: lanes 0-15 hold K={k,k+1}, lanes 16-31 K={k+2,k+3}
            v2f a;
            a.x = arow[k];
            a.y = arow[k + 1];
            // B fragment 4x16 from LDS: VGPR0 = row (k+2g), VGPR1 = row (k+2g+1)
            v2f b;
            b.x = sW[(k + 2 * g)     * HID + col];
            b.y = sW[(k + 2 * g + 1) * HID + col];
            acc = __builtin_amdgcn_wmma_f32_16x16x4_f32(
                /*neg_a=*/false, a, /*neg_b=*/false, b,
                /*c_mod=*/(short)0, acc, /*reuse_a=*/false, /*reuse_b=*/false);
        }

        // D layout: VGPR i -> M = rowBase + i + 8*g; N = col
        #pragma unroll
        for (int i = 0; i < 8; ++i) {
            int m = rowBase + i + 8 * g;
            float sm = rowScale ? rowScale[m] : 1.0f;
            C[(size_t)m * HID + col] = acc[i] * sm + badd;
        }
    }
}

// ---------------- edge scatter: agg[dst] += feat[src] (128 f32 per edge) ----
__global__ void rgcn_scatter(const float* __restrict__ feat,
                             const int* __restrict__ src, const int* __restrict__ dst,
                             float* __restrict__ agg, int n_edges) {
    int e    = blockIdx.x * (blockDim.x >> 5) + (threadIdx.x >> 5);
    int lane = threadIdx.x & 31;
    if (e >= n_edges) return;
    int s = src[e];
    int d = dst[e];
    float4 v = ((const float4*)(feat + (size_t)s * HID))[lane]; // global_load_b128
    float* o = agg + (size_t)d * HID + (size_t)lane * 4;
    unsafeAtomicAdd(o + 0, v.x);   // global_atomic_add_f32
    unsafeAtomicAdd(o + 1, v.y);
    unsafeAtomicAdd(o + 2, v.z);
    unsafeAtomicAdd(o + 3, v.w);
}

// ---------------- epilogue: h = (h * sIn[row] + bias[col]) [relu] ----------
__global__ void rgcn_finalize(float* __restrict__ h, const float* __restrict__ sIn,
                              const float* __restrict__ bias, long n, int relu) {
    long i = (long)blockIdx.x * blockDim.x + threadIdx.x;
    if (i < n) {
        int row = (int)(i >> 7);
        int col = (int)(i & (HID - 1));
        float v = h[i] * sIn[row] + bias[col];
        if (relu) v = fmaxf(v, 0.0f);
        h[i] = v;
    }
}

extern "C" void kernel_launch(void* const* d_in, const int* in_sizes, int n_in,
                              void* d_out, int out_size, void* d_ws, size_t ws_size,
                              hipStream_t stream) {
    const float* x_user  = (const float*)d_in[0];
    // d_in[1] (x_item), d_in[6] (W1_iu), d_in[7] (b1_iu) are dead in the reference
    const float* W_embed = (const float*)d_in[2];
    const float* b_embed = (const float*)d_in[3];
    const float* W1_ui   = (const float*)d_in[4];
    const float* b1_ui   = (const float*)d_in[5];
    const float* W2_iu   = (const float*)d_in[8];
    const float* b2_iu   = (const float*)d_in[9];
    const int*   src_ui  = (const int*)d_in[10];
    const int*   dst_ui  = (const int*)d_in[11];
    const int*   src_iu  = (const int*)d_in[12];
    const int*   dst_iu  = (const int*)d_in[13];
    float* out = (float*)d_out;

    const int n_user = in_sizes[0] / HID;   // 50000
    const int n_item = in_sizes[1] / HID;   // 50000
    const int n_edge = in_sizes[10];        // 500000

    const int uTiles = n_user / 16;         // 3125
    const int iTiles = n_item / 16;         // 3125
    const int gemmGrid = (uTiles < 1024) ? uTiles : 1024;

    // ---- workspace carve-up (floats) ----
    const long DEG_STRIDE = 50176;                 // >= 50000, 128-aligned
    float* ws      = (float*)d_ws;
    float* sOutUI  = ws + 0 * DEG_STRIDE;          // user out-degree (ui)
    float* sInUI   = ws + 1 * DEG_STRIDE;          // item in-degree (ui)
    float* sOutIU  = ws + 2 * DEG_STRIDE;          // item out-degree (iu)
    float* sInIU   = ws + 3 * DEG_STRIDE;          // user in-degree (iu)
    float* hu_agg  = ws + 4 * DEG_STRIDE;          // hu, later agg_item/h_item
    float* feat    = hu_agg + (long)n_user * HID;  // feat1 then feat2
    const long NODE_ELEMS = (long)n_user * HID;    // 6.4M floats

    // 1) zero degree counters, then histogram degrees, then rsqrt scales
    rgcn_zero_f32<<<512, 256, 0, stream>>>(sOutUI, 4 * DEG_STRIDE);
    rgcn_degrees<<<(n_edge + 255) / 256, 256, 0, stream>>>(
        src_ui, dst_ui, src_iu, dst_iu,
        (unsigned*)sOutUI, (unsigned*)sInUI, (unsigned*)sOutIU, (unsigned*)sInIU, n_edge);
    rgcn_scales<<<(n_user + 255) / 256, 256, 0, stream>>>(sOutUI, sInUI, sOutIU, sInIU, n_user);

    // 2) hu = x_user @ W_embed + b_embed
    rgcn_gemm_wmma<<<gemmGrid, 256, 0, stream>>>(
        x_user, W_embed, nullptr, b_embed, hu_agg, uTiles);

    // 3) feat1 = diag(sOutUI) * (hu @ W1_ui)
    rgcn_gemm_wmma<<<gemmGrid, 256, 0, stream>>>(
        hu_agg, W1_ui, sOutUI, nullptr, feat, uTiles);

    // 4) agg_item = segment_sum(feat1[src_ui], dst_ui)   (hu buffer reused)
    rgcn_zero_f32<<<2048, 256, 0, stream>>>(hu_agg, NODE_ELEMS);
    rgcn_scatter<<<(n_edge + 7) / 8, 256, 0, stream>>>(feat, src_ui, dst_ui, hu_agg, n_edge);

    // 5) h_item = relu(agg_item * sInUI + b1_ui)  (in place)
    rgcn_finalize<<<(int)((NODE_ELEMS + 255) / 256), 256, 0, stream>>>(
        hu_agg, sInUI, b1_ui, NODE_ELEMS, /*relu=*/1);

    // 6) feat2 = diag(sOutIU) * (h_item @ W2_iu)
    rgcn_gemm_wmma<<<gemmGrid, 256, 0, stream>>>(
        hu_agg, W2_iu, sOutIU, nullptr, feat, iTiles);

    // 7) out = segment_sum(feat2[src_iu], dst_iu)
    rgcn_zero_f32<<<2048, 256, 0, stream>>>(out, NODE_ELEMS);
    rgcn_scatter<<<(n_edge + 7) / 8, 256, 0, stream>>>(feat, src_iu, dst_iu, out, n_edge);

    // 8) out = out * sInIU + b2_iu (no relu on layer-2 output)
    rgcn_finalize<<<(int)((NODE_ELEMS + 255) / 256), 256, 0, stream>>>(
        out, sInIU, b2_iu, NODE_ELEMS, /*relu=*/0);
}